// DecoderWithSoftAttention_61907658604750
// MI455X (gfx1250) — compile-verified
//
#include <hip/hip_runtime.h>
#include <hip/hip_bf16.h>
#include <math.h>

// ---------------------------------------------------------------------------
// Problem constants (from the reference)
// ---------------------------------------------------------------------------
#define Bz   64
#define HW   14
#define P    196          // HW*HW
#define ENC  2048
#define Lc   20
#define Tz   19           // L - 1
#define Vz   10000
#define Ez   512
#define Dz   512
#define Az   512

typedef __attribute__((ext_vector_type(2))) float v2f;
typedef __attribute__((ext_vector_type(8))) float v8f;

__device__ __forceinline__ float sigf(float x) { return 1.0f / (1.0f + expf(-x)); }

// ---------------------------------------------------------------------------
// Kernel 0: stable descending argsort of caption lengths (B=64), emit
// sort_ind, decode_lengths, sorted captions (into d_out), sort_ind (d_out).
// ---------------------------------------------------------------------------
__global__ void sort_kernel(const int* __restrict__ cap_len,
                            const int* __restrict__ caps,
                            int* __restrict__ sort_ind_ws,
                            int* __restrict__ dec_len_ws,
                            int* __restrict__ caps_out,
                            int* __restrict__ sortind_out) {
  __shared__ int cl[Bz];
  int i = threadIdx.x;
  if (i < Bz) cl[i] = cap_len[i];
  __syncthreads();
  if (i < Bz) {
    int me = cl[i];
    int rank = 0;
    for (int j = 0; j < Bz; ++j) {
      int cj = cl[j];
      rank += (cj > me) || (cj == me && j < i);
    }
    sort_ind_ws[rank] = i;
    sortind_out[rank] = i;
    dec_len_ws[rank] = me - 1;
    for (int l = 0; l < Lc; ++l) caps_out[rank * Lc + l] = caps[i * Lc + l];
  }
}

// ---------------------------------------------------------------------------
// Kernel 1: mean over pixels of gathered encoder rows: mean(B,ENC)
// ---------------------------------------------------------------------------
__global__ void mean_kernel(const float* __restrict__ enc,
                            const int* __restrict__ sort_ind,
                            float* __restrict__ mean) {
  int i = blockIdx.x * blockDim.x + threadIdx.x;
  if (i >= Bz * ENC) return;
  int b = i >> 11;          // /ENC
  int e = i & (ENC - 1);
  const float* p = enc + (long)sort_ind[b] * P * ENC + e;
  float s = 0.0f;
  for (int k = 0; k < P; ++k) s += p[(long)k * ENC];
  mean[i] = s * (1.0f / (float)P);
}

// ---------------------------------------------------------------------------
// Generic fp32 WMMA GEMM:  C(M,N) = act( A(M,K) @ W(N,K)^T + bias ) [+C] [mask]
//   - wave tile 32x64: 2 A frags x 4 B frags -> 8 accumulators,
//     0.75 b64 loads per WMMA; double-buffered K loop (prefetch k+4)
//   - block = 8 waves (2 M x 4 N) -> 64x256 block tile
//   - A-row gather via rowmap (sort_ind) with rows_per_batch
//   - accumulate: C += result (for gates = x@W_ih^T + h@W_hh^T)
//   - act==1 : sigmoid
//   - dec_len!=null : zero rows where dec_len[row] <= t_mask (row == batch)
// Requires: M % 64 == 0, K % 4 == 0 and K >= 8, lda/ldw even.
// N tail: B column pointers clamp to N-1 (valid memory); out-of-range
// output columns are simply never stored (column n of D depends only on
// column n of B, so no contamination of valid columns).
// ---------------------------------------------------------------------------
__global__ void gemm_wmma_f32(const float* __restrict__ A, int lda,
                              const float* __restrict__ W, int ldw,
                              const float* __restrict__ bias,
                              float* __restrict__ C, long ldc,
                              int M, int N, int K,
                              const int* __restrict__ rowmap, int rows_per_batch,
                              int accumulate, int act,
                              const int* __restrict__ dec_len, int t_mask) {
  const int lane   = threadIdx.x & 31;
  const int wave   = threadIdx.x >> 5;
  const int wm     = wave >> 2;        // 0..1
  const int wn     = wave & 3;         // 0..3
  const int m0     = blockIdx.y * 64 + wm * 32;
  const int n0     = blockIdx.x * 256 + wn * 64;
  const int laneHi = lane >> 4;        // 0: K pair (0,1), 1: K pair (2,3)
  const int laneLo = lane & 15;

  // --- per-lane A element offsets (2 M sub-tiles), 32-bit arithmetic ------
  int offA[2];
  #pragma unroll
  for (int am = 0; am < 2; ++am) {
    int arow = m0 + am * 16 + laneLo;
    if (rowmap) {
      int b = arow / rows_per_batch;
      arow = rowmap[b] * rows_per_batch + (arow - b * rows_per_batch);
    }
    offA[am] = arow * lda + 2 * laneHi;
  }

  // --- per-lane B element offsets (4 N sub-tiles), clamped at N tail ------
  int offB[4];
  #pragma unroll
  for (int s = 0; s < 4; ++s) {
    int cn = n0 + 16 * s + laneLo;
    int cc = (cn < N) ? cn : (N - 1);
    offB[s] = cc * ldw + 2 * laneHi;
  }

  v8f acc[2][4];
  #pragma unroll
  for (int am = 0; am < 2; ++am)
    #pragma unroll
    for (int s = 0; s < 4; ++s) acc[am][s] = (v8f)(0.0f);

  // --- double-buffered K loop ---------------------------------------------
  v2f a0 = *(const v2f*)(A + offA[0]);
  v2f a1 = *(const v2f*)(A + offA[1]);
  v2f b0 = *(const v2f*)(W + offB[0]);
  v2f b1 = *(const v2f*)(W + offB[1]);
  v2f b2 = *(const v2f*)(W + offB[2]);
  v2f b3 = *(const v2f*)(W + offB[3]);

  #pragma unroll 2
  for (int k0 = 4; k0 < K; k0 += 4) {
    // prefetch next fragments
    v2f na0 = *(const v2f*)(A + offA[0] + k0);
    v2f na1 = *(const v2f*)(A + offA[1] + k0);
    v2f nb0 = *(const v2f*)(W + offB[0] + k0);
    v2f nb1 = *(const v2f*)(W + offB[1] + k0);
    v2f nb2 = *(const v2f*)(W + offB[2] + k0);
    v2f nb3 = *(const v2f*)(W + offB[3] + k0);

    acc[0][0] = __builtin_amdgcn_wmma_f32_16x16x4_f32(false, a0, false, b0, (short)0, acc[0][0], false, false);
    acc[0][1] = __builtin_amdgcn_wmma_f32_16x16x4_f32(false, a0, false, b1, (short)0, acc[0][1], false, false);
    acc[0][2] = __builtin_amdgcn_wmma_f32_16x16x4_f32(false, a0, false, b2, (short)0, acc[0][2], false, false);
    acc[0][3] = __builtin_amdgcn_wmma_f32_16x16x4_f32(false, a0, false, b3, (short)0, acc[0][3], false, false);
    acc[1][0] = __builtin_amdgcn_wmma_f32_16x16x4_f32(false, a1, false, b0, (short)0, acc[1][0], false, false);
    acc[1][1] = __builtin_amdgcn_wmma_f32_16x16x4_f32(false, a1, false, b1, (short)0, acc[1][1], false, false);
    acc[1][2] = __builtin_amdgcn_wmma_f32_16x16x4_f32(false, a1, false, b2, (short)0, acc[1][2], false, false);
    acc[1][3] = __builtin_amdgcn_wmma_f32_16x16x4_f32(false, a1, false, b3, (short)0, acc[1][3], false, false);

    a0 = na0; a1 = na1;
    b0 = nb0; b1 = nb1; b2 = nb2; b3 = nb3;
  }

  // final K group
  acc[0][0] = __builtin_amdgcn_wmma_f32_16x16x4_f32(false, a0, false, b0, (short)0, acc[0][0], false, false);
  acc[0][1] = __builtin_amdgcn_wmma_f32_16x16x4_f32(false, a0, false, b1, (short)0, acc[0][1], false, false);
  acc[0][2] = __builtin_amdgcn_wmma_f32_16x16x4_f32(false, a0, false, b2, (short)0, acc[0][2], false, false);
  acc[0][3] = __builtin_amdgcn_wmma_f32_16x16x4_f32(false, a0, false, b3, (short)0, acc[0][3], false, false);
  acc[1][0] = __builtin_amdgcn_wmma_f32_16x16x4_f32(false, a1, false, b0, (short)0, acc[1][0], false, false);
  acc[1][1] = __builtin_amdgcn_wmma_f32_16x16x4_f32(false, a1, false, b1, (short)0, acc[1][1], false, false);
  acc[1][2] = __builtin_amdgcn_wmma_f32_16x16x4_f32(false, a1, false, b2, (short)0, acc[1][2], false, false);
  acc[1][3] = __builtin_amdgcn_wmma_f32_16x16x4_f32(false, a1, false, b3, (short)0, acc[1][3], false, false);

  // --- epilogue: bias, accumulate, activation, row mask, store ------------
  #pragma unroll
  for (int am = 0; am < 2; ++am) {
    #pragma unroll
    for (int j = 0; j < 8; ++j) {
      int r = m0 + am * 16 + j + 8 * laneHi;   // output row
      float mk = 1.0f;
      if (dec_len) mk = (dec_len[r] > t_mask) ? 1.0f : 0.0f;
      #pragma unroll
      for (int s = 0; s < 4; ++s) {
        int cn = n0 + 16 * s + laneLo;
        if (cn < N) {
          long idx = (long)r * ldc + cn;
          float v = acc[am][s][j] + bias[cn];
          if (accumulate) v += C[idx];
          if (act == 1) v = sigf(v);
          C[idx] = v * mk;
        }
      }
    }
  }
}

// ---------------------------------------------------------------------------
// Attention kernel (one block per batch row):
//   e[p]   = sum_a tanh(att1[b,p,a] + att2[b,a]) * Wfull[a] + bfull
//   alpha  = softmax_p(e);  alphas_out[b,t,p] = mask ? alpha : 0
//   awe[b,e] = sum_p alpha[p] * enc[src(b),p,e]
// ---------------------------------------------------------------------------
__global__ void attention_kernel(const float* __restrict__ att1,
                                 const float* __restrict__ att2,
                                 const float* __restrict__ Wfull,
                                 const float* __restrict__ bfull,
                                 const float* __restrict__ enc,
                                 const int* __restrict__ sort_ind,
                                 const int* __restrict__ dec_len,
                                 int t,
                                 float* __restrict__ alphas_out,
                                 float* __restrict__ awe) {
  const int b   = blockIdx.x;
  const int tid = threadIdx.x;       // 256 threads
  __shared__ float att2s[Az];
  __shared__ float sh[256];
  __shared__ float alpha_s[P];

  for (int a = tid; a < Az; a += 256) att2s[a] = att2[b * Az + a];
  __syncthreads();

  const bool havep = (tid < P);
  float ev = 0.0f;
  if (havep) {
    const float* row = att1 + ((long)b * P + tid) * Az;
    float s = 0.0f;
    for (int a = 0; a < Az; ++a) s += tanhf(row[a] + att2s[a]) * Wfull[a];
    ev = s + bfull[0];
  }

  // softmax over p (block reduce max, then sum)
  sh[tid] = havep ? ev : -1e30f;
  __syncthreads();
  for (int off = 128; off > 0; off >>= 1) {
    if (tid < off) sh[tid] = fmaxf(sh[tid], sh[tid + off]);
    __syncthreads();
  }
  float mx = sh[0];
  __syncthreads();
  float ex = havep ? expf(ev - mx) : 0.0f;
  sh[tid] = ex;
  __syncthreads();
  for (int off = 128; off > 0; off >>= 1) {
    if (tid < off) sh[tid] += sh[tid + off];
    __syncthreads();
  }
  float sum = sh[0];
  __syncthreads();

  float al = ex / sum;
  if (havep) {
    alpha_s[tid] = al;
    float mk = (dec_len[b] > t) ? 1.0f : 0.0f;
    alphas_out[((long)b * Tz + t) * P + tid] = al * mk;
  }
  __syncthreads();

  const long base = (long)sort_ind[b] * P * ENC;
  for (int e = tid; e < ENC; e += 256) {
    float s = 0.0f;
    for (int p = 0; p < P; ++p) s += alpha_s[p] * enc[base + (long)p * ENC + e];
    awe[b * ENC + e] = s;
  }
}

// ---------------------------------------------------------------------------
// Build x = concat(emb[caps_sorted[:,t]], gate * awe)   -> (B, E+ENC)
// ---------------------------------------------------------------------------
__global__ void build_x_kernel(const float* __restrict__ emb,
                               const int* __restrict__ caps_sorted,
                               int t,
                               const float* __restrict__ gate,
                               const float* __restrict__ awe,
                               float* __restrict__ x) {
  int i = blockIdx.x * blockDim.x + threadIdx.x;
  const int XW = Ez + ENC;            // 2560
  if (i >= Bz * XW) return;
  int b = i / XW, k = i - b * XW;
  float v;
  if (k < Ez) {
    v = emb[(long)caps_sorted[b * Lc + t] * Ez + k];
  } else {
    int e = k - Ez;
    v = gate[b * ENC + e] * awe[b * ENC + e];
  }
  x[i] = v;
}

// ---------------------------------------------------------------------------
// LSTM element-wise: gates(B,4D) split [i,f,g,o]; masked h/c update.
// ---------------------------------------------------------------------------
__global__ void lstm_kernel(const float* __restrict__ gates,
                            const int* __restrict__ dec_len,
                            int t,
                            float* __restrict__ h, float* __restrict__ c,
                            float* __restrict__ h_new, float* __restrict__ c_new) {
  int i = blockIdx.x * blockDim.x + threadIdx.x;
  if (i >= Bz * Dz) return;
  int b = i >> 9;                     // /Dz
  int d = i & (Dz - 1);
  const float* g = gates + (long)b * 4 * Dz;
  float ig = sigf(g[d]);
  float fg = sigf(g[Dz + d]);
  float gg = tanhf(g[2 * Dz + d]);
  float og = sigf(g[3 * Dz + d]);
  float cn = fg * c[i] + ig * gg;
  float hn = og * tanhf(cn);
  h_new[i] = hn;
  c_new[i] = cn;
  if (dec_len[b] > t) { h[i] = hn; c[i] = cn; }
}

// ---------------------------------------------------------------------------
// Host-side launcher
// ---------------------------------------------------------------------------
static inline void launch_gemm(hipStream_t s,
                               const float* A, int lda,
                               const float* W, int ldw,
                               const float* bias,
                               float* C, long ldc,
                               int M, int N, int K,
                               const int* rowmap, int rpb,
                               int accumulate, int act,
                               const int* dec_len, int t_mask) {
  dim3 grid((N + 255) / 256, (M + 63) / 64);
  gemm_wmma_f32<<<grid, 256, 0, s>>>(A, lda, W, ldw, bias, C, ldc, M, N, K,
                                     rowmap, rpb, accumulate, act, dec_len, t_mask);
}

extern "C" void kernel_launch(void* const* d_in, const int* in_sizes, int n_in,
                              void* d_out, int out_size, void* d_ws, size_t ws_size,
                              hipStream_t stream) {
  // ---- inputs (setup_inputs order) ----
  const float* encoder_out = (const float*)d_in[0];   // (B,14,14,ENC)
  const int*   enc_caps    = (const int*)  d_in[1];   // (B,L)
  const int*   cap_len     = (const int*)  d_in[2];   // (B,1)
  const float* emb         = (const float*)d_in[3];   // (V,E)
  const float* W_ih        = (const float*)d_in[4];   // (4D, E+ENC)
  const float* b_ih        = (const float*)d_in[5];
  const float* W_hh        = (const float*)d_in[6];   // (4D, D)
  const float* b_hh        = (const float*)d_in[7];
  const float* W_init_h    = (const float*)d_in[8];   // (D, ENC)
  const float* b_init_h    = (const float*)d_in[9];
  const float* W_init_c    = (const float*)d_in[10];
  const float* b_init_c    = (const float*)d_in[11];
  const float* W_fbeta     = (const float*)d_in[12];  // (ENC, D)
  const float* b_fbeta     = (const float*)d_in[13];
  const float* W_fc        = (const float*)d_in[14];  // (V, D)
  const float* b_fc        = (const float*)d_in[15];
  const float* W_enc_att   = (const float*)d_in[16];  // (A, ENC)
  const float* b_enc_att   = (const float*)d_in[17];
  const float* W_dec_att   = (const float*)d_in[18];  // (A, D)
  const float* b_dec_att   = (const float*)d_in[19];
  const float* W_full_att  = (const float*)d_in[20];  // (1, A)
  const float* b_full_att  = (const float*)d_in[21];  // (1,)

  // ---- outputs: predictions(B,T,V) f32 | caps(B,L) i32 | alphas(B,T,P) f32 | sort_ind(B) i32
  float* pred_out    = (float*)d_out;
  int*   caps_out    = (int*)((float*)d_out + (long)Bz * Tz * Vz);
  float* alphas_out  = (float*)((float*)d_out + (long)Bz * Tz * Vz + (long)Bz * Lc);
  int*   sortind_out = (int*)((float*)d_out + (long)Bz * Tz * Vz + (long)Bz * Lc + (long)Bz * Tz * P);

  // ---- workspace carve-up ----
  char* ws = (char*)d_ws;
  size_t off = 0;
  auto carve = [&](size_t bytes) { void* p = ws + off; off = (off + bytes + 255) & ~((size_t)255); return p; };
  int*   sort_ind = (int*)  carve(Bz * sizeof(int));
  int*   dec_len  = (int*)  carve(Bz * sizeof(int));
  float* mean     = (float*)carve((size_t)Bz * ENC * sizeof(float));
  float* h        = (float*)carve((size_t)Bz * Dz  * sizeof(float));
  float* c        = (float*)carve((size_t)Bz * Dz  * sizeof(float));
  float* h_new    = (float*)carve((size_t)Bz * Dz  * sizeof(float));
  float* c_new    = (float*)carve((size_t)Bz * Dz  * sizeof(float));
  float* att2     = (float*)carve((size_t)Bz * Az  * sizeof(float));
  float* awe      = (float*)carve((size_t)Bz * ENC * sizeof(float));
  float* gate     = (float*)carve((size_t)Bz * ENC * sizeof(float));
  float* xbuf     = (float*)carve((size_t)Bz * (Ez + ENC) * sizeof(float));
  float* gates    = (float*)carve((size_t)Bz * 4 * Dz * sizeof(float));
  float* att1     = (float*)carve((size_t)Bz * P * Az * sizeof(float));
  (void)ws_size; (void)in_sizes; (void)n_in; (void)out_size;

  // 0) sort + emit caps_sorted / sort_ind outputs
  sort_kernel<<<1, 64, 0, stream>>>(cap_len, enc_caps, sort_ind, dec_len,
                                    caps_out, sortind_out);

  // 1) mean over pixels (gathered)
  mean_kernel<<<(Bz * ENC + 255) / 256, 256, 0, stream>>>(encoder_out, sort_ind, mean);

  // 2) h0 / c0
  launch_gemm(stream, mean, ENC, W_init_h, ENC, b_init_h, h, Dz,
              Bz, Dz, ENC, nullptr, 1, 0, 0, nullptr, 0);
  launch_gemm(stream, mean, ENC, W_init_c, ENC, b_init_c, c, Dz,
              Bz, Dz, ENC, nullptr, 1, 0, 0, nullptr, 0);

  // 3) att1 = enc_sorted @ W_enc_att^T + b   (M = B*P rows, gathered)
  launch_gemm(stream, encoder_out, ENC, W_enc_att, ENC, b_enc_att, att1, Az,
              Bz * P, Az, ENC, sort_ind, P, 0, 0, nullptr, 0);

  // 4) recurrence
  for (int t = 0; t < Tz; ++t) {
    // att2 = h @ W_dec_att^T + b
    launch_gemm(stream, h, Dz, W_dec_att, Dz, b_dec_att, att2, Az,
                Bz, Az, Dz, nullptr, 1, 0, 0, nullptr, 0);

    // attention: e, softmax, alphas(out, masked), awe
    attention_kernel<<<Bz, 256, 0, stream>>>(att1, att2, W_full_att, b_full_att,
                                             encoder_out, sort_ind, dec_len, t,
                                             alphas_out, awe);

    // gate = sigmoid(h @ W_fbeta^T + b)
    launch_gemm(stream, h, Dz, W_fbeta, Dz, b_fbeta, gate, ENC,
                Bz, ENC, Dz, nullptr, 1, 0, 1, nullptr, 0);

    // x = [emb_t | gate*awe]
    build_x_kernel<<<(Bz * (Ez + ENC) + 255) / 256, 256, 0, stream>>>(
        emb, caps_out, t, gate, awe, xbuf);

    // gates = x @ W_ih^T + b_ih ; gates += h @ W_hh^T + b_hh
    launch_gemm(stream, xbuf, Ez + ENC, W_ih, Ez + ENC, b_ih, gates, 4 * Dz,
                Bz, 4 * Dz, Ez + ENC, nullptr, 1, 0, 0, nullptr, 0);
    launch_gemm(stream, h, Dz, W_hh, Dz, b_hh, gates, 4 * Dz,
                Bz, 4 * Dz, Dz, nullptr, 1, 1, 0, nullptr, 0);

    // LSTM cell + masked h/c update
    lstm_kernel<<<(Bz * Dz + 255) / 256, 256, 0, stream>>>(
        gates, dec_len, t, h, c, h_new, c_new);

    // preds = h_new @ W_fc^T + b_fc, masked, written into predictions[:, t, :]
    launch_gemm(stream, h_new, Dz, W_fc, Dz, b_fc,
                pred_out + (long)t * Vz, (long)Tz * Vz,
                Bz, Vz, Dz, nullptr, 1, 0, 0, dec_len, t);
  }
}